// PEGH_40570261078754
// MI455X (gfx1250) — compile-verified
//
#include <hip/hip_runtime.h>

#define DIMC   512
#define GRIDS  224
#define CELLS  (GRIDS * GRIDS)

typedef float v2f __attribute__((ext_vector_type(2)));
typedef float v8f __attribute__((ext_vector_type(8)));

// ws layout: [0..15] mins(2 int)+pad | [64 ..) zero row (DIMC f32) | map (CELLS int) | s (n f32)
#define WS_ZROW_OFF 64
#define WS_MAP_OFF  (WS_ZROW_OFF + DIMC * 4)
#define WS_S_OFF    (WS_MAP_OFF + CELLS * 4)

// ---------------------------------------------------------------- init: map=-1, mins=INT_MAX, zrow=0
__global__ void k_init(int* __restrict__ mins, int* __restrict__ map,
                       float* __restrict__ zrow, int cells) {
    int i = blockIdx.x * blockDim.x + threadIdx.x;
    if (i < cells) map[i] = -1;
    if (i < DIMC)  zrow[i] = 0.0f;
    if (i == 0) { mins[0] = 0x7fffffff; mins[1] = 0x7fffffff; }
}

// ---------------------------------------------------------------- mins over pos
__global__ void k_min(const int* __restrict__ pos, int* __restrict__ mins, int n) {
    int i = blockIdx.x * blockDim.x + threadIdx.x;
    if (i < n) {
        atomicMin(&mins[0], pos[2 * i]);
        atomicMin(&mins[1], pos[2 * i + 1]);
    }
}

// ---------------------------------------------------------------- scatter token ids into map
__global__ void k_scatter(const int* __restrict__ pos, const int* __restrict__ mins,
                          int* __restrict__ map, int n) {
    int i = blockIdx.x * blockDim.x + threadIdx.x;
    if (i < n) {
        int r = pos[2 * i]     - mins[0];
        int c = pos[2 * i + 1] - mins[1];
        map[r * GRIDS + c] = i;
    }
}

// ---------------------------------------------------------------- per-token channel sum (mask)
__global__ void k_rowsum(const float* __restrict__ x, float* __restrict__ s, int n) {
    int lane = threadIdx.x & 31;
    int tok  = blockIdx.x * (blockDim.x >> 5) + (threadIdx.x >> 5);
    if (tok >= n) return;
    float v = 0.0f;
    for (int c = lane; c < DIMC; c += 32) v += x[(size_t)tok * DIMC + c];
    #pragma unroll
    for (int o = 16; o > 0; o >>= 1) v += __shfl_xor(v, o, 32);
    if (lane == 0) s[tok] = v;
}

// ---------------------------------------------------------------- main: WMMA depthwise conv
// block = 256 threads = 8 waves; 16 tokens x 128 channels per block; grid = (N/16, 512/128)
__global__ __launch_bounds__(256) void k_conv(
    const float* __restrict__ x, const float* __restrict__ w, const float* __restrict__ bia,
    const int* __restrict__ pos, const int* __restrict__ mins, const int* __restrict__ map,
    const float* __restrict__ s, const float* __restrict__ zrow,
    float* __restrict__ out, int n)
{
    __shared__ int   nbr[16][9];
    __shared__ float msk[16];

    const int t0  = blockIdx.x * 16;
    const int tid = threadIdx.x;

    // stage 9 neighbor token-ids per token (shared by all 8 waves of this block)
    if (tid < 144) {
        int tt  = t0 + tid / 9;
        int tap = tid % 9;                 // tap = ky*3+kx ; dc = tap/3-1, dr = tap%3-1
        int id  = -1;
        if (tt < n) {
            int r  = pos[2 * tt]     - mins[0];
            int c  = pos[2 * tt + 1] - mins[1];
            int rr = r + (tap % 3) - 1;
            int cc = c + (tap / 3) - 1;
            if (rr >= 0 && rr < GRIDS && cc >= 0 && cc < GRIDS)
                id = map[rr * GRIDS + cc];
        }
        nbr[tid / 9][tid % 9] = id;
    }
    if (tid < 16) {
        int tt = t0 + tid;
        msk[tid] = (tt < n && s[tt] != 0.0f) ? 1.0f : 0.0f;
    }
    __syncthreads();

    const int wave  = tid >> 5;
    const int lane  = tid & 31;
    const int m     = lane & 15;           // A row (token) / B,C column (channel)
    const int khalf = lane >> 4;           // K half for A/B fragments
    const int c0    = blockIdx.y * 128 + wave * 16;

    // per-lane weight column: wv[tap] = w[(c0+m)*9 + tap]
    float wv[9];
    #pragma unroll
    for (int t = 0; t < 9; ++t) wv[t] = w[(size_t)(c0 + m) * 9 + t];

    v8f acc = {};
    #pragma unroll
    for (int tap = 0; tap < 9; ++tap) {
        int nid = nbr[m][tap];
        // missing neighbor -> point at a zeroed 512-float row: loads return exact 0,
        // no mask multiply / pointer clamp needed on the WMMA feed path
        const float* row = (nid >= 0) ? (x + (size_t)nid * DIMC + c0) : zrow;
        #pragma unroll
        for (int k4 = 0; k4 < 4; ++k4) {
            int ch = 4 * k4 + 2 * khalf;               // K index base for this lane
            v2f a;                                     // A[m, ch], A[m, ch+1]
            a.x = row[ch];
            a.y = row[ch + 1];
            v2f bb;                                    // block-diagonal B: B[k, n]=w iff n==4*k4+k
            bb.x = (m == ch)     ? wv[tap] : 0.0f;
            bb.y = (m == ch + 1) ? wv[tap] : 0.0f;
            acc = __builtin_amdgcn_wmma_f32_16x16x4_f32(
                /*neg_a=*/false, a, /*neg_b=*/false, bb,
                /*c_mod=*/(short)0, acc, /*reuse_a=*/false, /*reuse_b=*/false);
        }
    }

    // epilogue: D[v] = out for token (t0 + v + 8*khalf), channel (c0 + m)
    const float bv = bia[c0 + m];
    #pragma unroll
    for (int v = 0; v < 8; ++v) {
        int tl = v + 8 * khalf;
        int tg = t0 + tl;
        if (tg < n) {
            size_t idx = (size_t)tg * DIMC + c0 + m;
            out[idx] = x[idx] + msk[tl] * (acc[v] + bv);
        }
    }
}

// ---------------------------------------------------------------- launcher
extern "C" void kernel_launch(void* const* d_in, const int* in_sizes, int n_in,
                              void* d_out, int out_size, void* d_ws, size_t ws_size,
                              hipStream_t stream) {
    const float* x   = (const float*)d_in[0];
    const int*   pos = (const int*)  d_in[1];
    const float* w   = (const float*)d_in[2];
    const float* b   = (const float*)d_in[3];
    float*       out = (float*)d_out;
    const int    n   = in_sizes[1] / 2;

    char*  wsb  = (char*)d_ws;
    int*   mins = (int*)wsb;
    float* zrow = (float*)(wsb + WS_ZROW_OFF);
    int*   map  = (int*)(wsb + WS_MAP_OFF);
    float* s    = (float*)(wsb + WS_S_OFF);

    k_init   <<<(CELLS + 255) / 256, 256, 0, stream>>>(mins, map, zrow, CELLS);
    k_min    <<<(n + 255) / 256,     256, 0, stream>>>(pos, mins, n);
    k_scatter<<<(n + 255) / 256,     256, 0, stream>>>(pos, mins, map, n);
    k_rowsum <<<(n + 7) / 8,         256, 0, stream>>>(x, s, n);

    dim3 grid((n + 15) / 16, DIMC / 128);
    k_conv<<<grid, 256, 0, stream>>>(x, w, b, pos, mins, map, s, zrow, out, n);
}